// DWT_59863254172102
// MI455X (gfx1250) — compile-verified
//
#include <hip/hip_runtime.h>

typedef __attribute__((ext_vector_type(2))) float v2f;
typedef __attribute__((ext_vector_type(8))) float v8f;
typedef int v4i __attribute__((vector_size(4 * sizeof(int))));
typedef __attribute__((ext_vector_type(4))) unsigned int u32x4;
typedef __attribute__((ext_vector_type(8))) int i32x8;
typedef __attribute__((ext_vector_type(4))) int i32x4;

// ---- CDNA5 async global->LDS path (probe-verified round 1/2) ----
#if defined(__has_builtin)
#  if __has_builtin(__builtin_amdgcn_global_load_async_to_lds_b32) && \
      __has_builtin(__builtin_amdgcn_global_load_async_to_lds_b128)
#    define USE_ASYNC_LDS 1
#  endif
#  if __has_builtin(__builtin_amdgcn_tensor_load_to_lds) && \
      __has_builtin(__builtin_amdgcn_s_wait_tensorcnt)
#    define USE_TDM 1
#  endif
#endif
#ifndef USE_ASYNC_LDS
#  define USE_ASYNC_LDS 0
#endif
#ifndef USE_TDM
#  define USE_TDM 0
#endif

#if defined(__has_builtin)
#  if __has_builtin(__builtin_amdgcn_s_wait_asynccnt)
#    define WAIT_ASYNC() __builtin_amdgcn_s_wait_asynccnt(0)
#  endif
#endif
#ifndef WAIT_ASYNC
#  define WAIT_ASYNC() asm volatile("s_wait_asynccnt 0" ::: "memory")
#endif

// async builtin param types (from hipcc diagnostics, round 1):
//   b32 : (int AS1*, int AS3*, imm offset, imm cpol)
//   b128: (v4i AS1*, v4i AS3*, imm offset, imm cpol)
typedef __attribute__((address_space(1))) int  GI32;
typedef __attribute__((address_space(3))) int  LI32;
typedef __attribute__((address_space(1))) v4i  GI128;
typedef __attribute__((address_space(3))) v4i  LI128;

// LOW taps [0..7], HIGH taps [8..15]
__constant__ float FTAB16[16] = {
     0.1629f,  0.5055f,  0.4464f, -0.0198f, -0.1323f,  0.0218f,  0.0233f, -0.0075f,
    -0.0075f, -0.0233f,  0.0218f,  0.1323f, -0.0198f, -0.4464f,  0.5055f, -0.1629f
};

// One wave (32 lanes) per block. Each block: one (batch,channel) plane,
// one 16x16 output tile, all 4 subbands.
__global__ __launch_bounds__(32) void dwt59_kernel(const float* __restrict__ x,
                                                   float* __restrict__ out) {
    constexpr int H = 256, W = 256, C = 64, HO = 125, WO = 125;
    constexpr int IN_ROWS = 38;   // 2*15 + 8 input rows per tile
    constexpr int IN_PITCH = 40;  // padded to 10 float4 / row

    const int lane = threadIdx.x;            // 0..31
    const int tx = blockIdx.x & 7;           // 8 x-tiles
    const int ty = blockIdx.x >> 3;          // 8 y-tiles
    const int ch = blockIdx.y;               // 64 channels
    const int b  = blockIdx.z;               // 16 batch
    const int y0 = ty * 16, x0 = tx * 16;
    const int row0 = 2 * y0, col0 = 2 * x0;  // col0 multiple of 32 -> 128B aligned

    __shared__ float in_lds[IN_ROWS * IN_PITCH];  // 6080 B
    __shared__ float t_lds[IN_ROWS * 32];         // [row][x(16)*2 + p], 4864 B
    __shared__ float ftab[16];

    const float* plane = x + (size_t)(b * C + ch) * (H * W);

    if (lane < 16) ftab[lane] = FTAB16[lane];

    // ---------------- stage 38x40 input tile into LDS ----------------
    const bool interior = (tx < 7) && (ty < 7);  // no clamping needed
    if (interior) {
#if USE_TDM
        // Tensor Data Mover: one DMA for the whole 38x40-f32 2D tile.
        const float* gp0 = plane + row0 * W + col0;
        unsigned long long ga = (unsigned long long)(const void*)gp0;
        unsigned lds_base = (unsigned)(unsigned long long)(LI32*)&in_lds[0];
        unsigned td0 = (unsigned)(W - col0);   // remaining extent dim0 (>= tile_dim0)
        unsigned td1 = (unsigned)(H - row0);   // remaining extent dim1 (>= tile_dim1)
        // D# group0: count=1 | lds_addr | global_addr[56:0] | type=2
        u32x4 g0;
        g0.x = 1u;                                   // count=1, user flags 0
        g0.y = lds_base;                             // lds_addr
        g0.z = (unsigned)(ga & 0xFFFFFFFFu);         // global_addr[31:0]
        g0.w = (unsigned)((ga >> 32) & 0x01FFFFFFu) | (2u << 30);  // addr[56:32], type=2
        // D# group1: data_size=4B, tensor_dim0/1, tile 40x38, stride 256
        i32x8 g1;
        g1[0] = (int)(2u << 16);                          // data_size=2 (4 bytes)
        g1[1] = (int)((td0 & 0xFFFFu) << 16);             // tensor_dim0[15:0] @ [63:48]
        g1[2] = (int)((td0 >> 16) | ((td1 & 0xFFFFu) << 16));  // td0[31:16], td1[15:0]
        g1[3] = (int)((td1 >> 16) | (40u << 16));         // td1[31:16], tile_dim0=40
        g1[4] = (int)38u;                                 // tile_dim1=38, tile_dim2=0
        g1[5] = (int)256u;                                // tensor_dim0_stride[31:0]
        g1[6] = 0;                                        // stride0[47:32], stride1[15:0]
        g1[7] = 0;                                        // stride1[47:16]
        i32x4 gz4 = {0, 0, 0, 0};                         // groups 2/3 unused (2D)
        i32x8 gz8 = {0, 0, 0, 0, 0, 0, 0, 0};            // extra arg of 6-arg form
        __builtin_amdgcn_tensor_load_to_lds(g0, g1, gz4, gz4, gz8, 0);
        __builtin_amdgcn_s_wait_tensorcnt(0);
#elif USE_ASYNC_LDS
        for (int j = lane; j < IN_ROWS * 10; j += 32) {
            int r  = j / 10;
            int c4 = (j % 10) * 4;
            const float* gp = plane + (row0 + r) * W + col0 + c4;
            float* lp = &in_lds[r * IN_PITCH + c4];
            __builtin_amdgcn_global_load_async_to_lds_b128(
                (GI128*)const_cast<float*>(gp), (LI128*)lp, 0, 0);
        }
        WAIT_ASYNC();
#else
        for (int j = lane; j < IN_ROWS * 10; j += 32) {
            int r  = j / 10;
            int c4 = (j % 10) * 4;
            const float4 v = *(const float4*)(plane + (row0 + r) * W + col0 + c4);
            *(float4*)&in_lds[r * IN_PITCH + c4] = v;
        }
#endif
    } else {
#if USE_ASYNC_LDS
        for (int i = lane; i < IN_ROWS * IN_PITCH; i += 32) {
            int r  = i / IN_PITCH;
            int cc = i % IN_PITCH;
            int gr = min(row0 + r, H - 1);   // clamped; OOB rows feed discarded outputs
            int gc = min(col0 + cc, W - 1);
            const float* gp = plane + gr * W + gc;
            float* lp = &in_lds[i];
            __builtin_amdgcn_global_load_async_to_lds_b32(
                (GI32*)const_cast<float*>(gp), (LI32*)lp, 0, 0);
        }
        WAIT_ASYNC();
#else
        for (int i = lane; i < IN_ROWS * IN_PITCH; i += 32) {
            int r  = i / IN_PITCH;
            int cc = i % IN_PITCH;
            int gr = min(row0 + r, H - 1);
            int gc = min(col0 + cc, W - 1);
            in_lds[i] = plane[gr * W + gc];
        }
#endif
    }
    // single wave: LDS ops are in-order within the wave; no barrier needed.

    // ---------------- horizontal pass: t[r][x][p] = sum_k in[r][2x+k]*f_p[k] ----
    const int p_h = lane & 1;       // horizontal subband for this lane
    const int xc  = lane >> 1;      // output column 0..15
    float fcoef[8];
#pragma unroll
    for (int k = 0; k < 8; ++k) fcoef[k] = ftab[p_h * 8 + k];

    for (int r = 0; r < IN_ROWS; ++r) {
        const float* src = &in_lds[r * IN_PITCH + 2 * xc];
        float acc = 0.f;
#pragma unroll
        for (int k = 0; k < 8; ++k) acc = fmaf(src[k], fcoef[k], acc);
        t_lds[r * 32 + lane] = acc;  // = r*32 + xc*2 + p_h
    }

    // ---------------- vertical pass via banded-matrix WMMA ----------------
    // D[m=(q<<3)|ys][n] = sum_k A[m][k] * B[k][n],  A[m][k] = f_q[k-2*ys]
    // A: 16x24 (6 chunks of K=4), constant; built once per wave.
    const int mA    = lane & 15;
    const int qA    = mA >> 3;
    const int ysA   = mA & 7;
    const int khalf = (lane < 16) ? 0 : 2;  // K layout: lanes0-15 hold K={0,1}, 16-31 K={2,3}
    const int n     = lane & 15;

    v2f A[6];
#pragma unroll
    for (int cch = 0; cch < 6; ++cch) {
        int k0 = 4 * cch + khalf;
        int j0 = k0 - 2 * ysA;
        int j1 = j0 + 1;
        float a0 = ftab[qA * 8 + min(max(j0, 0), 7)];
        float a1 = ftab[qA * 8 + min(max(j1, 0), 7)];
        A[cch].x = (j0 >= 0 && j0 < 8) ? a0 : 0.f;
        A[cch].y = (j1 >= 0 && j1 < 8) ? a1 : 0.f;
    }

#pragma unroll
    for (int pp = 0; pp < 2; ++pp) {          // horizontal subband of B
#pragma unroll
        for (int yb = 0; yb < 2; ++yb) {      // 8-row output block
            const int rbase = 16 * yb;
            v8f d = {0.f, 0.f, 0.f, 0.f, 0.f, 0.f, 0.f, 0.f};
#pragma unroll
            for (int cch = 0; cch < 6; ++cch) {
                int k0 = 4 * cch + khalf;
                int r0 = min(rbase + k0,     IN_ROWS - 1);  // rows past band have A==0
                int r1 = min(rbase + k0 + 1, IN_ROWS - 1);
                v2f bb;
                bb.x = t_lds[r0 * 32 + n * 2 + pp];
                bb.y = t_lds[r1 * 32 + n * 2 + pp];
                d = __builtin_amdgcn_wmma_f32_16x16x4_f32(
                        false, A[cch], false, bb, (short)0, d, false, false);
            }
            // D layout: VGPR i -> M = i + (lane<16 ? 0 : 8), N = lane&15
#pragma unroll
            for (int i = 0; i < 8; ++i) {
                int M   = i + ((lane < 16) ? 0 : 8);
                int qq  = M >> 3;       // vertical subband
                int yss = M & 7;
                int yg  = y0 + 8 * yb + yss;
                int xg  = x0 + n;
                if (yg < HO && xg < WO) {
                    int fsub = 2 * qq + pp;  // (ll,lh,hl,hh) = vertical-major
                    // 64M elements total -> 32-bit indexing (avoids v_mul_u64)
                    unsigned idx =
                        (((unsigned)(b * (4 * C) + fsub * C + ch) * HO + (unsigned)yg) * WO) +
                        (unsigned)xg;
                    out[idx] = d[i];
                }
            }
        }
    }
}

extern "C" void kernel_launch(void* const* d_in, const int* in_sizes, int n_in,
                              void* d_out, int out_size, void* d_ws, size_t ws_size,
                              hipStream_t stream) {
    (void)in_sizes; (void)n_in; (void)d_ws; (void)ws_size; (void)out_size;
    const float* x = (const float*)d_in[0];
    float* out = (float*)d_out;
    dim3 grid(64, 64, 16);   // (8x8 tiles, 64 channels, 16 batch)
    dim3 block(32);          // one wave32
    hipLaunchKernelGGL(dwt59_kernel, grid, block, 0, stream, x, out);
}